// AttentionDist_88691074662802
// MI455X (gfx1250) — compile-verified
//
#include <hip/hip_runtime.h>
#include <math.h>

typedef __attribute__((ext_vector_type(2))) float v2f;
typedef __attribute__((ext_vector_type(8))) float v8f;

#define DIM 256
#define CLD 32
#define NCLS 21
#define H 128
#define B 8
#define N 128
#define CS 256

// CDNA5 async global->LDS copy path (ASYNCcnt). Guarded: toolchains differ in
// what they expose here; fallback is plain staged loads (still correct).
#if defined(__gfx1250__) && \
    __has_builtin(__builtin_amdgcn_global_load_async_to_lds_b128) && \
    __has_builtin(__builtin_amdgcn_s_wait_asynccnt)
#define HAVE_ASYNC_LDS 1
#define AS1 __attribute__((address_space(1)))
#define AS3 __attribute__((address_space(3)))
typedef int v4i_ __attribute__((vector_size(16)));   // matches builtin param type
#endif

// ---------------------------------------------------------------------------
// Kernel 1: q_pre = queries @ Wq                      (1024 x 128, K=256)
//           ctx_pre = context @ Wc + le @ Wl + b1     (2048 x 128, K=256 + 32)
// One 16-row tile per block; the A tile (16x256, 16KB) is staged once in LDS
// via async global->LDS (all 8 waves share it); 8 waves each own one 16-col
// tile of H=128 and run a V_WMMA_F32_16X16X4_F32 K-loop.
// ---------------------------------------------------------------------------
__global__ void prep_gemm(const float* __restrict__ queries,   // (B*N, DIM)
                          const float* __restrict__ context,   // (B*CS, DIM)
                          const int*   __restrict__ labels,    // (B*CS)
                          const float* __restrict__ label_emb, // (NCLS, CLD)
                          const float* __restrict__ W1,        // (2*DIM+CLD, H)
                          const float* __restrict__ b1,        // (H)
                          float* __restrict__ q_pre,           // (B*N, H)
                          float* __restrict__ ctx_pre)         // (B*CS, H)
{
    __shared__ __align__(16) float s_a[16 * DIM];   // 16 KB shared A tile

    const int lane = threadIdx.x & 31;
    const int wave = threadIdx.x >> 5;     // 0..7 -> column tile of H
    const int rt   = blockIdx.x;           // 0..191 row tile
    const int col0 = wave * 16;
    const bool hi  = lane >= 16;
    const int  l16 = lane & 15;

    const bool isQ = rt < (B * N / 16);                 // first 64 tiles: queries
    const int  row0 = isQ ? rt * 16 : (rt - B * N / 16) * 16;
    const float* A = isQ ? queries : context;
    const float* W = W1 + (isQ ? 0 : DIM) * H;          // Wq or Wc

    // ---- stage A tile (16 contiguous rows = 16*256 floats) into LDS ----
    const float* Abase = A + (size_t)row0 * DIM;
#if HAVE_ASYNC_LDS
#pragma unroll
    for (int j = 0; j < 4; ++j) {                       // 1024 x 16B chunks
        const int c = threadIdx.x + 256 * j;
        __builtin_amdgcn_global_load_async_to_lds_b128(
            (AS1 v4i_*)(Abase + c * 4),
            (AS3 v4i_*)(s_a + c * 4), 0, 0);
    }
    __builtin_amdgcn_s_wait_asynccnt(0);
#else
    for (int i = threadIdx.x; i < 16 * DIM; i += 256) s_a[i] = Abase[i];
#endif
    __syncthreads();

    v8f acc = {};
    for (int k0 = 0; k0 < DIM; k0 += 4) {
        const int ka = k0 + (hi ? 2 : 0);
        // A tile 16x4 from LDS: lane l16 holds row M=l16, (K=ka,K=ka+1)
        v2f a = *(const v2f*)(s_a + l16 * DIM + ka);
        // B tile 4x16: vgpr0 = row K=ka (across lanes), vgpr1 = row K=ka+1
        v2f bv;
        bv.x = W[(size_t)(ka + 0) * H + col0 + l16];
        bv.y = W[(size_t)(ka + 1) * H + col0 + l16];
        acc = __builtin_amdgcn_wmma_f32_16x16x4_f32(false, a, false, bv,
                                                    (short)0, acc, false, false);
    }

    if (!isQ) {
        // label-embedding contribution: A[m][k] = label_emb[label[row m]][k]
        const float* Wl = W1 + (size_t)(2 * DIM) * H;
        const int lab = labels[row0 + l16];
        const float* le = label_emb + (size_t)lab * CLD;
        for (int k0 = 0; k0 < CLD; k0 += 4) {
            const int ka = k0 + (hi ? 2 : 0);
            v2f a; a.x = le[ka]; a.y = le[ka + 1];
            v2f bv;
            bv.x = Wl[(size_t)(ka + 0) * H + col0 + l16];
            bv.y = Wl[(size_t)(ka + 1) * H + col0 + l16];
            acc = __builtin_amdgcn_wmma_f32_16x16x4_f32(false, a, false, bv,
                                                        (short)0, acc, false, false);
        }
        const float bias = b1[col0 + l16];
#pragma unroll
        for (int i = 0; i < 8; ++i) acc[i] += bias;
    }

    float* out = isQ ? q_pre : ctx_pre;
    const int col = col0 + l16;
#pragma unroll
    for (int i = 0; i < 8; ++i) {
        const int m = i + (hi ? 8 : 0);   // D layout: vgpr i -> M=i (lo) / i+8 (hi)
        out[(size_t)(row0 + m) * H + col] = acc[i];
    }
}

// ---------------------------------------------------------------------------
// Kernel 2: fused attention tail for one (batch, 16-query-row) tile.
//   attn[m][c] = sum_h relu(q_pre[m][h] + ctx_pre[c][h]) * W2[h] + b2
//   l_softmax = attn - lse(attn);  P = exp(l_softmax)
//   context_vector = P @ context[b]            (WMMA f32 16x16x4)
//   l_softmax_classes[m][k] = logsumexp_{c: label==k} attn[m][c] - lse[m]
// ---------------------------------------------------------------------------
__global__ void attn_tail(const float* __restrict__ q_pre,    // (B*N, H)
                          const float* __restrict__ ctx_pre,  // (B*CS, H)
                          const float* __restrict__ context,  // (B*CS, DIM)
                          const int*   __restrict__ labels,   // (B*CS)
                          const float* __restrict__ W2,       // (H)
                          const float* __restrict__ b2,       // (1)
                          float* __restrict__ out_cv,         // (B*N, DIM)
                          float* __restrict__ out_lsc,        // (B*N, NCLS)
                          float* __restrict__ out_attn)       // (B*N, CS)
{
    __shared__ __align__(16) float s_qp[16 * H];     // 8 KB
    __shared__ __align__(16) float s_w2[H];          // 512 B
    __shared__ __align__(16) float s_attn[16 * CS];  // 16 KB, later holds P
    __shared__ float s_lse[16];

    const int t  = threadIdx.x;                      // 0..255
    const int b  = blockIdx.x >> 3;                  // 8 n-tiles per batch
    const int nt = blockIdx.x & 7;
    const int row_base = b * N + nt * 16;            // query-row base
    const int ctx_base = b * CS;                     // context-row base

    // stage q_pre tile (16 contiguous rows) + W2 into LDS
    const float* qbase = q_pre + (size_t)row_base * H;
#if HAVE_ASYNC_LDS
#pragma unroll
    for (int j = 0; j < 2; ++j) {                    // 512 x 16B chunks
        const int c = t + 256 * j;
        __builtin_amdgcn_global_load_async_to_lds_b128(
            (AS1 v4i_*)(qbase + c * 4),
            (AS3 v4i_*)(s_qp + c * 4), 0, 0);
    }
    if (t < 32)                                      // 128 floats of W2
        __builtin_amdgcn_global_load_async_to_lds_b128(
            (AS1 v4i_*)(W2 + t * 4),
            (AS3 v4i_*)(s_w2 + t * 4), 0, 0);
    __builtin_amdgcn_s_wait_asynccnt(0);
#else
    for (int i = t; i < 16 * H; i += 256) s_qp[i] = qbase[i];
    if (t < H) s_w2[t] = W2[t];
#endif

    // prefetch context[b] (256 KB, consumed by phase-5 WMMA B-loads) so the
    // fetch overlaps the VALU-heavy phases 1-4: global_prefetch_b8 per 256B.
    {
        const float* cbase = context + (size_t)ctx_base * DIM;
        for (int c = t; c < (CS * DIM * 4) / 256; c += 256)   // 1024 lines
            __builtin_prefetch(cbase + c * 64, 0, 1);
    }
    __syncthreads();

    // ---- Phase 1: relu-dot. thread t owns context column c = t. ----
    const float4* cp4 = (const float4*)(ctx_pre + (size_t)(ctx_base + t) * H);
    float acc[16];
#pragma unroll
    for (int m = 0; m < 16; ++m) acc[m] = 0.f;

    for (int h = 0; h < H; h += 4) {
        const float4 c4 = cp4[h >> 2];
        const float4 w4 = *(const float4*)&s_w2[h];
#pragma unroll
        for (int m = 0; m < 16; ++m) {
            const float4 q4 = *(const float4*)&s_qp[m * H + h];
            acc[m] += fmaxf(q4.x + c4.x, 0.f) * w4.x
                    + fmaxf(q4.y + c4.y, 0.f) * w4.y
                    + fmaxf(q4.z + c4.z, 0.f) * w4.z
                    + fmaxf(q4.w + c4.w, 0.f) * w4.w;
        }
    }
    const float bias2 = b2[0];
#pragma unroll
    for (int m = 0; m < 16; ++m) {
        const float v = acc[m] + bias2;
        s_attn[m * CS + t] = v;
        out_attn[(size_t)(row_base + m) * CS + t] = v;
    }
    __syncthreads();

    // ---- Phase 2: per-row logsumexp over CS=256 ----
    if (t < 16) {
        float mx = -INFINITY;
        for (int c = 0; c < CS; ++c) mx = fmaxf(mx, s_attn[t * CS + c]);
        float s = 0.f;
        for (int c = 0; c < CS; ++c) s += expf(s_attn[t * CS + c] - mx);
        s_lse[t] = mx + logf(s);
    }
    __syncthreads();

    // ---- Phase 3: per-class streaming logsumexp (16 rows x 21 classes) ----
    for (int idx = t; idx < 16 * NCLS; idx += 256) {
        const int m = idx & 15;
        const int k = idx >> 4;
        float mrun = -INFINITY, s = 0.f;
        for (int c = 0; c < CS; ++c) {
            if (labels[ctx_base + c] == k) {
                const float x = s_attn[m * CS + c];
                if (x > mrun) { s = s * expf(mrun - x) + 1.f; mrun = x; }
                else          { s += expf(x - mrun); }
            }
        }
        out_lsc[(size_t)(row_base + m) * NCLS + k] =
            (s > 0.f) ? (mrun + logf(s) - s_lse[m]) : -INFINITY;
    }
    __syncthreads();

    // ---- Phase 4: P = exp(attn - lse) (overwrite s_attn) ----
#pragma unroll
    for (int m = 0; m < 16; ++m)
        s_attn[m * CS + t] = expf(s_attn[m * CS + t] - s_lse[m]);
    __syncthreads();

    // ---- Phase 5: context_vector = P(16x256) @ context[b](256x256), WMMA ----
    const int lane = t & 31;
    const int wave = t >> 5;
    const bool hi  = lane >= 16;
    const int  l16 = lane & 15;

    for (int ct = 0; ct < 2; ++ct) {                  // 8 waves x 2 = 16 col tiles
        const int col0 = (wave * 2 + ct) * 16;
        v8f dacc = {};
        for (int k0 = 0; k0 < CS; k0 += 4) {
            const int ka = k0 + (hi ? 2 : 0);
            v2f a;                                    // A from LDS (P tile)
            a.x = s_attn[l16 * CS + ka];
            a.y = s_attn[l16 * CS + ka + 1];
            v2f bv;                                   // B from global (context)
            bv.x = context[(size_t)(ctx_base + ka + 0) * DIM + col0 + l16];
            bv.y = context[(size_t)(ctx_base + ka + 1) * DIM + col0 + l16];
            dacc = __builtin_amdgcn_wmma_f32_16x16x4_f32(false, a, false, bv,
                                                         (short)0, dacc, false, false);
        }
#pragma unroll
        for (int i = 0; i < 8; ++i) {
            const int m = i + (hi ? 8 : 0);
            out_cv[(size_t)(row_base + m) * DIM + col0 + l16] = dacc[i];
        }
    }
}

// ---------------------------------------------------------------------------
extern "C" void kernel_launch(void* const* d_in, const int* in_sizes, int n_in,
                              void* d_out, int out_size, void* d_ws, size_t ws_size,
                              hipStream_t stream) {
    const float* queries   = (const float*)d_in[0];  // (B,N,DIM)
    const float* context   = (const float*)d_in[1];  // (B,CS,DIM)
    const int*   labels    = (const int*)  d_in[2];  // (B,CS)
    /* d_in[3] = mask: unused by the reference computation */
    const float* label_emb = (const float*)d_in[4];  // (NCLS,CLD)
    const float* W1        = (const float*)d_in[5];  // (2*DIM+CLD, H)
    const float* b1        = (const float*)d_in[6];  // (H)
    const float* W2        = (const float*)d_in[7];  // (H,1)
    const float* b2        = (const float*)d_in[8];  // (1)

    float* q_pre   = (float*)d_ws;                        // B*N*H   floats
    float* ctx_pre = q_pre + (size_t)B * N * H;           // B*CS*H  floats

    float* out      = (float*)d_out;
    float* out_cv   = out;                                // B*N*DIM
    float* out_lsc  = out_cv + (size_t)B * N * DIM;       // B*N*NCLS
    float* out_attn = out_lsc + (size_t)B * N * NCLS;     // B*N*CS

    // 64 query row-tiles + 128 context row-tiles; 8 waves = 8 col tiles of H
    prep_gemm<<<(B * N + B * CS) / 16, 256, 0, stream>>>(
        queries, context, labels, label_emb, W1, b1, q_pre, ctx_pre);

    // one block per (batch, 16-query-row) tile
    attn_tail<<<B * (N / 16), 256, 0, stream>>>(
        q_pre, ctx_pre, context, labels, W2, b2, out_cv, out_lsc, out_attn);
}